// PDNBlock_36850819400184
// MI455X (gfx1250) — compile-verified
//
#include <hip/hip_runtime.h>

#define Nn     100000
#define Ee     1600000
#define IN_C   128
#define OUT_C  64
#define E_DIM  16
#define E_HID  32
#define EPSn   1e-5f

typedef __attribute__((ext_vector_type(2))) float v2f;
typedef __attribute__((ext_vector_type(4))) float v4f;
typedef __attribute__((ext_vector_type(8))) float v8f;

// ---------------- K0: init output (scatter accumulator), deg (=1.0 self-loop), stats ----
__global__ void k_init(float* __restrict__ out, float* __restrict__ deg,
                       float* __restrict__ stats) {
  int i = blockIdx.x * 256 + threadIdx.x;          // grid covers Nn*OUT_C exactly
  out[i] = 0.0f;
  if (i < Nn)  deg[i] = 1.0f;                      // self-loop weight pre-added
  if (i < 128) stats[i] = 0.0f;
}

// ---------------- K1: edge MLP gate + weighted-degree accumulation ---------------------
__global__ void k_edge_mlp(const float* __restrict__ ea, const long long* __restrict__ ei,
                           const float* __restrict__ w1, const float* __restrict__ b1,
                           const float* __restrict__ w2, const float* __restrict__ b2,
                           float* __restrict__ ew, float* __restrict__ deg) {
  int e = blockIdx.x * 256 + threadIdx.x;          // grid covers Ee exactly
  float a[E_DIM];
  const v4f* p = (const v4f*)(ea + e * E_DIM);
#pragma unroll
  for (int t = 0; t < 4; t++) {
    v4f q = p[t];
    a[4*t] = q.x; a[4*t+1] = q.y; a[4*t+2] = q.z; a[4*t+3] = q.w;
  }
  float acc = b2[0];
#pragma unroll
  for (int j = 0; j < E_HID; j++) {
    float h = b1[j];
#pragma unroll
    for (int k = 0; k < E_DIM; k++) h = fmaf(a[k], w1[j*E_DIM + k], h);  // uniform weights -> s_load
    h = fmaxf(h, 0.0f);
    acc = fmaf(h, w2[j], acc);
  }
  float g = 1.0f / (1.0f + __expf(-acc));          // sigmoid
  ew[e] = g;
  atomicAdd(&deg[(int)ei[Ee + e]], g);             // weighted degree on targets
}

// ---------------- K2: deg -> D^{-1/2} in place ----------------------------------------
__global__ void k_dinv(float* __restrict__ deg) {
  int i = blockIdx.x * 256 + threadIdx.x;
  if (i < Nn) {
    float d = deg[i];
    deg[i] = (d > 0.0f) ? rsqrtf(d) : 0.0f;
  }
}

// ---------------- K3: xt = x @ lin_w.T via V_WMMA_F32_16X16X4_F32 ---------------------
// One wave per 16-row M tile; 4 N-tiles of 16; K=128 in steps of 4.
// lin_w [64,128] staged in LDS once per block; B[k][n] = lin_w[n][k].
__global__ void __launch_bounds__(256) k_xt_wmma(const float* __restrict__ x,
                                                 const float* __restrict__ lw,
                                                 float* __restrict__ xt) {
  __shared__ float w[OUT_C * IN_C];                // 32 KB
  for (int i = threadIdx.x; i < (OUT_C * IN_C) / 4; i += 256)
    ((v4f*)w)[i] = ((const v4f*)lw)[i];
  __syncthreads();

  int wave = threadIdx.x >> 5;
  int lane = threadIdx.x & 31;
  int tile = blockIdx.x * 8 + wave;
  if (tile >= Nn / 16) return;                     // whole-wave exit; EXEC stays full

  int l16  = lane & 15;
  int koff = (lane >> 4) * 2;                      // lanes 16-31 hold K pair {2,3}
  const float* xrow = x + (tile * 16 + l16) * IN_C;

  v8f acc0 = {}, acc1 = {}, acc2 = {}, acc3 = {};
#pragma unroll 4
  for (int k0 = 0; k0 < IN_C; k0 += 4) {
    v2f a  = *(const v2f*)(xrow + k0 + koff);                       // global_load_b64
    v2f b0 = *(const v2f*)(w + ( 0 + l16) * IN_C + k0 + koff);      // ds_load_b64
    v2f b1 = *(const v2f*)(w + (16 + l16) * IN_C + k0 + koff);
    v2f b2 = *(const v2f*)(w + (32 + l16) * IN_C + k0 + koff);
    v2f b3 = *(const v2f*)(w + (48 + l16) * IN_C + k0 + koff);
    acc0 = __builtin_amdgcn_wmma_f32_16x16x4_f32(false, a, false, b0, (short)0, acc0, false, false);
    acc1 = __builtin_amdgcn_wmma_f32_16x16x4_f32(false, a, false, b1, (short)0, acc1, false, false);
    acc2 = __builtin_amdgcn_wmma_f32_16x16x4_f32(false, a, false, b2, (short)0, acc2, false, false);
    acc3 = __builtin_amdgcn_wmma_f32_16x16x4_f32(false, a, false, b3, (short)0, acc3, false, false);
  }

  // C/D layout: VGPR j -> M = j (+8 for lanes 16-31), N = lane%16
  int rbase = tile * 16 + (lane >> 4) * 8;
#pragma unroll
  for (int j = 0; j < 8; j++) {
    float* o = xt + (rbase + j) * OUT_C + l16;
    o[0]  = acc0[j];
    o[16] = acc1[j];
    o[32] = acc2[j];
    o[48] = acc3[j];
  }
}

// ---------------- K4: gather xt[row], scale by norm, scatter-add to out[col] ----------
__global__ void k_scatter(const long long* __restrict__ ei, const float* __restrict__ ew,
                          const float* __restrict__ dinv, const float* __restrict__ xt,
                          float* __restrict__ out) {
  int idx = blockIdx.x * 256 + threadIdx.x;        // grid covers Ee*64 exactly
  int e = idx >> 6, c = idx & 63;
  int r   = (int)ei[e];
  int col = (int)ei[Ee + e];
  float nrm = dinv[r] * ew[e] * dinv[col];
  atomicAdd(out + col * OUT_C + c, nrm * xt[r * OUT_C + c]);
}

// ---------------- K5: + self-loop + bias, PReLU, fused sum/sumsq reduction ------------
__global__ void k_post1(float* __restrict__ out, const float* __restrict__ xt,
                        const float* __restrict__ dinv, const float* __restrict__ bias,
                        const float* __restrict__ prelu_a, float* __restrict__ stats) {
  __shared__ float ls[128];
  if (threadIdx.x < 128) ls[threadIdx.x] = 0.0f;
  __syncthreads();

  int idx = blockIdx.x * 256 + threadIdx.x;        // grid covers Nn*OUT_C exactly
  int i = idx >> 6, c = idx & 63;
  float di = dinv[i];
  float v = out[idx] + (di * di) * xt[idx] + bias[c];  // self-loop norm = 1/deg
  float pa = prelu_a[0];
  v = (v >= 0.0f) ? v : pa * v;                    // PReLU
  out[idx] = v;
  atomicAdd(&ls[c], v);                            // ds_add_f32
  atomicAdd(&ls[64 + c], v * v);
  __syncthreads();
  if (threadIdx.x < 128) atomicAdd(&stats[threadIdx.x], ls[threadIdx.x]);
}

// ---------------- K6: per-channel mean/var -> (s*mean, weight*rstd) -------------------
__global__ void k_stats(const float* __restrict__ stats, const float* __restrict__ ms,
                        const float* __restrict__ gw, float* __restrict__ der) {
  int c = threadIdx.x;
  if (c < OUT_C) {
    float mean = stats[c]      * (1.0f / (float)Nn);
    float m2   = stats[64 + c] * (1.0f / (float)Nn);
    float s = ms[c];
    // E[(v - s*mean)^2] = E[v^2] - 2*s*mean^2 + s^2*mean^2
    float var = m2 - 2.0f * s * mean * mean + s * s * mean * mean;
    der[c]      = s * mean;
    der[64 + c] = gw[c] * rsqrtf(var + EPSn);
  }
}

// ---------------- K7: final normalize -------------------------------------------------
__global__ void k_post2(float* __restrict__ out, const float* __restrict__ der,
                        const float* __restrict__ gb) {
  int idx = blockIdx.x * 256 + threadIdx.x;        // grid covers Nn*OUT_C exactly
  int c = idx & 63;
  out[idx] = der[64 + c] * (out[idx] - der[c]) + gb[c];
}

extern "C" void kernel_launch(void* const* d_in, const int* in_sizes, int n_in,
                              void* d_out, int out_size, void* d_ws, size_t ws_size,
                              hipStream_t stream) {
  const float*     x    = (const float*)d_in[0];
  const long long* ei   = (const long long*)d_in[1];   // int64 edge_index [2,E]
  const float*     ea   = (const float*)d_in[2];
  const float*     lw   = (const float*)d_in[3];
  const float*     bias = (const float*)d_in[4];
  const float*     w1   = (const float*)d_in[5];
  const float*     b1   = (const float*)d_in[6];
  const float*     w2   = (const float*)d_in[7];
  const float*     b2   = (const float*)d_in[8];
  const float*     pa   = (const float*)d_in[9];
  const float*     gw   = (const float*)d_in[10];
  const float*     gb   = (const float*)d_in[11];
  const float*     gms  = (const float*)d_in[12];
  float* out = (float*)d_out;

  float* ws    = (float*)d_ws;
  float* ew    = ws;                           // E
  float* deg   = ws + Ee;                      // N (becomes dinv in place)
  float* xt    = ws + Ee + Nn;                 // N*64
  float* stats = ws + Ee + Nn + Nn * OUT_C;    // 128 (sum | sumsq)
  float* der   = stats + 128;                  // 128 (s*mean | scale)

  k_init    <<<(Nn * OUT_C) / 256, 256, 0, stream>>>(out, deg, stats);
  k_edge_mlp<<<Ee / 256,           256, 0, stream>>>(ea, ei, w1, b1, w2, b2, ew, deg);
  k_dinv    <<<(Nn + 255) / 256,   256, 0, stream>>>(deg);
  k_xt_wmma <<<(Nn / 16 + 7) / 8,  256, 0, stream>>>(x, lw, xt);
  k_scatter <<<(Ee * 64) / 256,    256, 0, stream>>>(ei, ew, deg, xt, out);
  k_post1   <<<(Nn * OUT_C) / 256, 256, 0, stream>>>(out, xt, deg, bias, pa, stats);
  k_stats   <<<1, 64,                   0, stream>>>(stats, gms, gw, der);
  k_post2   <<<(Nn * OUT_C) / 256, 256, 0, stream>>>(out, der, gb);
}